// Net_66434554135156
// MI455X (gfx1250) — compile-verified
//
#include <hip/hip_runtime.h>
#include <hip/hip_bf16.h>

// GraphSAGE on MI455X (gfx1250, wave32).
// Roofline: GEMMs ~20 GFLOP fp32 (trivial for WMMA); edge scatter ~5 GB but
// h (51MB) and neigh (51MB) fit in 192MB L2 -> scatter bound at L2/atomic rate.
// => fp32 everywhere, V_WMMA_F32_16X16X4_F32 for all GEMMs.
// Round 4: input-GEMM k-loop peeled to an even 28-iteration main loop so the
// compiler's unroll-by-2 has a single bottom exit (removes the 32-reg
// accumulator copy block seen in round 3).

#define N_NODES 100000
#define N_EDGES 1600000
#define DIM 128
#define KIN 117
#define ND (N_NODES * DIM)

typedef __attribute__((ext_vector_type(2))) float v2f;
typedef __attribute__((ext_vector_type(8))) float v8f;

// ---------------------------------------------------------------------------
// Degree: deg[d] += 1 per edge (fp32 hardware atomics)
// ---------------------------------------------------------------------------
__global__ void degree_kernel(const int* __restrict__ dst, float* __restrict__ deg) {
    int e = blockIdx.x * blockDim.x + threadIdx.x;
    if (e < N_EDGES) {
        unsafeAtomicAdd(&deg[dst[e]], 1.0f);
    }
}

// ---------------------------------------------------------------------------
// Edge scatter: neigh[dst] += h[src]; one wave (32 lanes) per edge, 4 floats/lane
// ---------------------------------------------------------------------------
__global__ void scatter_kernel(const float* __restrict__ h,
                               const int* __restrict__ src,
                               const int* __restrict__ dst,
                               float* __restrict__ neigh) {
    int gid = blockIdx.x * blockDim.x + threadIdx.x;
    int e = gid >> 5;
    int q = gid & 31;
    if (e >= N_EDGES) return;
    int s = src[e];
    int d = dst[e];
    const float4 v = ((const float4*)(h + (size_t)s * DIM))[q];
    float* np = neigh + (size_t)d * DIM + q * 4;
    unsafeAtomicAdd(np + 0, v.x);
    unsafeAtomicAdd(np + 1, v.y);
    unsafeAtomicAdd(np + 2, v.z);
    unsafeAtomicAdd(np + 3, v.w);
}

// ---------------------------------------------------------------------------
// Input layer: out = tanh(h0[100000,117] @ W_in[117,128] + b_in)
// Block = 256 threads = 8 waves arranged 4(M) x 2(N): block tile 64 x 128.
// Each wave computes a 16x64 tile: 4 WMMA v8f accumulators (32 VGPRs).
// A frag (16x4 f32): lane holds A[M=lane%16, K=(lane/16)*2 + {0,1}]
// B frag (4x16 f32): lane holds B[K=(lane/16)*2 + {0,1}, N=lane%16]
// C/D (16x16 f32):   VGPR r, lane L -> M = r + 8*(L/16), N = L%16
// ---------------------------------------------------------------------------
__global__ void input_gemm_tanh(const float* __restrict__ h0,
                                const float* __restrict__ Win,
                                const float* __restrict__ bin,
                                float* __restrict__ out) {
    const int wave  = threadIdx.x >> 5;     // 0..7
    const int waveM = wave >> 1;            // 0..3
    const int waveN = wave & 1;             // 0..1
    const int lane  = threadIdx.x & 31;
    const int mrow  = lane & 15;
    const int kh    = (lane >> 4) * 2;      // 0 or 2
    const int rowBase = blockIdx.x * 64 + waveM * 16;
    const int colBase = waveN * 64;
    int row  = rowBase + mrow;
    int rowC = row < N_NODES ? row : (N_NODES - 1);   // clamped for safe loads
    const float* arow = h0 + (size_t)rowC * KIN;      // 4B-aligned only

    v8f acc[4] = {};

    // One unguarded k-step (ka and ka+1 are < 117 by construction).
    auto kstep = [&](int ka) {
        v2f a;
        a.x = arow[ka];
        a.y = arow[ka + 1];
        const float* w0 = Win + ka * DIM + colBase;
        const float* w1 = w0 + DIM;
#pragma unroll
        for (int t = 0; t < 4; ++t) {
            const int n = t * 16 + mrow;
            v2f b;
            b.x = w0[n];
            b.y = w1[n];
            acc[t] = __builtin_amdgcn_wmma_f32_16x16x4_f32(
                false, a, false, b, (short)0, acc[t], false, false);
        }
    };

    // Main loop: 28 iterations (even -> clean unroll-by-2, single bottom exit).
    for (int k0 = 0; k0 < 112; k0 += 4)
        kstep(k0 + kh);
    // Peeled iteration k0 = 112: ka+1 <= 115 < 117, still unguarded.
    kstep(112 + kh);

    // Tail: k0 = 116 -> ka in {116, 118}; only 116 valid.
    // Branchless: unconditional clamped-address load + select-to-zero.
    {
        const int ka  = 116 + kh;
        const bool ok = (ka < KIN);
        const int kaC = ok ? ka : 0;
        float ax = arow[kaC];
        v2f a;
        a.x = ok ? ax : 0.0f;
        a.y = 0.0f;
        const float* w0 = Win + kaC * DIM + colBase;
#pragma unroll
        for (int t = 0; t < 4; ++t) {
            const int n = t * 16 + mrow;
            float bx = w0[n];
            v2f b;
            b.x = ok ? bx : 0.0f;
            b.y = 0.0f;
            acc[t] = __builtin_amdgcn_wmma_f32_16x16x4_f32(
                false, a, false, b, (short)0, acc[t], false, false);
        }
    }

    // Epilogue: bias + tanh. Uniform fast path for full tiles.
    const int rhalf = 8 * (lane >> 4);
    if (rowBase + 16 <= N_NODES) {
#pragma unroll
        for (int t = 0; t < 4; ++t) {
            const int n = colBase + t * 16 + mrow;
            const float bias = bin[n];
#pragma unroll
            for (int r = 0; r < 8; ++r) {
                const int orow = rowBase + r + rhalf;
                out[(size_t)orow * DIM + n] = tanhf(acc[t][r] + bias);
            }
        }
    } else {
#pragma unroll
        for (int t = 0; t < 4; ++t) {
            const int n = colBase + t * 16 + mrow;
            const float bias = bin[n];
#pragma unroll
            for (int r = 0; r < 8; ++r) {
                const int orow = rowBase + r + rhalf;
                if (orow < N_NODES)
                    out[(size_t)orow * DIM + n] = tanhf(acc[t][r] + bias);
            }
        }
    }
}

// ---------------------------------------------------------------------------
// SAGE layer: out = relu(h @ Wself + (neigh/max(deg,1)) @ Wneigh + b)
// Fused: both GEMMs accumulate into the same 16x64 tile per wave. K=128
// (32 even iterations), fully unguarded, branch-free k-loop.
// ---------------------------------------------------------------------------
__global__ void sage_gemm(const float* __restrict__ hin,
                          const float* __restrict__ neigh,
                          const float* __restrict__ deg,
                          const float* __restrict__ Wself,
                          const float* __restrict__ Wneigh,
                          const float* __restrict__ bias,
                          float* __restrict__ out) {
    const int wave  = threadIdx.x >> 5;
    const int waveM = wave >> 1;
    const int waveN = wave & 1;
    const int lane  = threadIdx.x & 31;
    const int mrow  = lane & 15;
    const int kh    = (lane >> 4) * 2;
    const int rowBase = blockIdx.x * 64 + waveM * 16;
    const int colBase = waveN * 64;
    int row  = rowBase + mrow;
    int rowC = row < N_NODES ? row : (N_NODES - 1);

    const float dv   = deg[rowC];
    const float invd = 1.0f / fmaxf(dv, 1.0f);
    const float* hrow = hin   + (size_t)rowC * DIM;   // 512B-aligned rows
    const float* nrow = neigh + (size_t)rowC * DIM;

    v8f acc[4] = {};

    for (int k0 = 0; k0 < DIM; k0 += 4) {
        const int ka = k0 + kh;                 // even -> 8B-aligned v2f loads
        v2f aH = *(const v2f*)(hrow + ka);
        v2f aN = *(const v2f*)(nrow + ka);
        aN.x *= invd;
        aN.y *= invd;
        const float* ws0 = Wself  + ka * DIM + colBase;
        const float* ws1 = ws0 + DIM;
        const float* wn0 = Wneigh + ka * DIM + colBase;
        const float* wn1 = wn0 + DIM;
#pragma unroll
        for (int t = 0; t < 4; ++t) {
            const int n = t * 16 + mrow;
            v2f bS, bN;
            bS.x = ws0[n];
            bS.y = ws1[n];
            bN.x = wn0[n];
            bN.y = wn1[n];
            acc[t] = __builtin_amdgcn_wmma_f32_16x16x4_f32(
                false, aH, false, bS, (short)0, acc[t], false, false);
            acc[t] = __builtin_amdgcn_wmma_f32_16x16x4_f32(
                false, aN, false, bN, (short)0, acc[t], false, false);
        }
    }

    // Epilogue: bias + relu. Uniform fast path for full tiles.
    const int rhalf = 8 * (lane >> 4);
    if (rowBase + 16 <= N_NODES) {
#pragma unroll
        for (int t = 0; t < 4; ++t) {
            const int n = colBase + t * 16 + mrow;
            const float bv = bias[n];
#pragma unroll
            for (int r = 0; r < 8; ++r) {
                const int orow = rowBase + r + rhalf;
                out[(size_t)orow * DIM + n] = fmaxf(acc[t][r] + bv, 0.0f);
            }
        }
    } else {
#pragma unroll
        for (int t = 0; t < 4; ++t) {
            const int n = colBase + t * 16 + mrow;
            const float bv = bias[n];
#pragma unroll
            for (int r = 0; r < 8; ++r) {
                const int orow = rowBase + r + rhalf;
                if (orow < N_NODES)
                    out[(size_t)orow * DIM + n] = fmaxf(acc[t][r] + bv, 0.0f);
            }
        }
    }
}

// ---------------------------------------------------------------------------
extern "C" void kernel_launch(void* const* d_in, const int* in_sizes, int n_in,
                              void* d_out, int out_size, void* d_ws, size_t ws_size,
                              hipStream_t stream) {
    const float* h0      = (const float*)d_in[0];
    const int*   src     = (const int*)d_in[1];
    const int*   dst     = (const int*)d_in[2];
    const float* W_in    = (const float*)d_in[3];
    const float* b_in    = (const float*)d_in[4];
    const float* W_self  = (const float*)d_in[5];   // [3,128,128]
    const float* W_neigh = (const float*)d_in[6];   // [3,128,128]
    const float* b_sage  = (const float*)d_in[7];   // [3,128]
    float* out = (float*)d_out;

    float* hA    = (float*)d_ws;          // N*128
    float* hB    = hA + ND;               // N*128
    float* neigh = hB + ND;               // N*128
    float* degb  = neigh + ND;            // N

    const int gemmBlocks = (N_NODES + 63) / 64;     // 1563 (64 rows per block)

    // Degree (constant across layers)
    hipMemsetAsync(degb, 0, N_NODES * sizeof(float), stream);
    degree_kernel<<<(N_EDGES + 255) / 256, 256, 0, stream>>>(dst, degb);

    // Input projection + tanh
    input_gemm_tanh<<<gemmBlocks, 256, 0, stream>>>(h0, W_in, b_in, hA);

    float* cur = hA;
    float* nxt = hB;
    for (int l = 0; l < 3; ++l) {
        hipMemsetAsync(neigh, 0, (size_t)ND * sizeof(float), stream);
        scatter_kernel<<<(N_EDGES * 32 + 255) / 256, 256, 0, stream>>>(cur, src, dst, neigh);
        float* o = (l == 2) ? out : nxt;
        sage_gemm<<<gemmBlocks, 256, 0, stream>>>(
            cur, neigh, degb,
            W_self + (size_t)l * DIM * DIM,
            W_neigh + (size_t)l * DIM * DIM,
            b_sage + (size_t)l * DIM,
            o);
        float* tmp = cur; cur = o; nxt = tmp;
    }
}